// NET_OnePointSpectra_64441689309403
// MI455X (gfx1250) — compile-verified
//
#include <hip/hip_runtime.h>

typedef __attribute__((ext_vector_type(2))) float v2f;
typedef __attribute__((ext_vector_type(8))) float v8f;

#define NK1 64
#define NK2 192
#define NK3 192
#define NT2 12   // 12 tiles of 16 rows of i2

// D = A(16x4 f32) * B(4x16 f32) + C(16x16 f32), exact f32 WMMA.
__device__ __forceinline__ v8f wmma4(v2f a, v2f b, v8f c) {
  return __builtin_amdgcn_wmma_f32_16x16x4_f32(
      /*neg_a=*/false, a, /*neg_b=*/false, b,
      /*c_mod=*/(short)0, c, /*reuse_a=*/false, /*reuse_b=*/false);
}

__device__ __forceinline__ float lx16(float x) { return __shfl_xor(x, 16, 32); }

__device__ __forceinline__ v8f relu8(v8f x) {
#pragma unroll
  for (int r = 0; r < 8; ++r) x[r] = fmaxf(x[r], 0.f);
  return x;
}

struct MLP {
  v2f a0;        // layer0 weights (Wᵀ, K padded 3->4 with zeros) in A layout
  v2f a1[4];     // layer1 weights, 4 K-chunks in A layout
  v2f a2[4];     // layer2 weights
  v8f cb0, cb1, cb2;  // biases broadcast in C/D layout (c[r] = b[half*8+r])
  float w3v[8];  // final 16->1 weights, per-lane slice
  float b3s;
};

// One 16-feature layer: hin is previous D (feat = half*8+r striped over lanes' VGPRs,
// point = lane%16). Build B chunks (feat = 4c+2*half+v) via a single half-swap shuffle.
__device__ __forceinline__ v8f layer16(const v2f aw[4], v8f hin, v8f cbias, int Hh) {
  float d[8], ds[8];
#pragma unroll
  for (int r = 0; r < 8; ++r) { d[r] = hin[r]; ds[r] = lx16(d[r]); }
  v8f acc = cbias;
#pragma unroll
  for (int c = 0; c < 4; ++c) {
    const int base = (c & 1) * 4;
    v2f b;
    if (c < 2) {  // source features live in half 0
      b.x = Hh ? ds[base + 2] : d[base + 0];
      b.y = Hh ? ds[base + 3] : d[base + 1];
    } else {      // source features live in half 1
      b.x = Hh ? d[base + 2] : ds[base + 0];
      b.y = Hh ? d[base + 3] : ds[base + 1];
    }
    acc = wmma4(aw[c], b, acc);
  }
  return acc;
}

// Full MLP for one 16-point tile; returns mlp_out replicated in both lane halves.
__device__ __forceinline__ float mlp_tile(v2f b0op, const MLP& P, int Hh) {
  v8f h = relu8(wmma4(P.a0, b0op, P.cb0));
  h = relu8(layer16(P.a1, h, P.cb1, Hh));
  h = relu8(layer16(P.a2, h, P.cb2, Hh));
  float p = 0.f;
#pragma unroll
  for (int r = 0; r < 8; ++r) p += h[r] * P.w3v[r];
  return p + lx16(p) + P.b3s;
}

__global__ void __launch_bounds__(256) ops_phase1(
    const float* __restrict__ k1, const float* __restrict__ k2,
    const float* __restrict__ k3, const float* __restrict__ logL,
    const float* __restrict__ logGamma, const float* __restrict__ logM,
    const float* __restrict__ W0, const float* __restrict__ b0,
    const float* __restrict__ W1, const float* __restrict__ b1,
    const float* __restrict__ W2, const float* __restrict__ b2,
    const float* __restrict__ W3, const float* __restrict__ b3,
    float* __restrict__ partial) {
  const int i1 = blockIdx.x;
  const int tile = blockIdx.y;
  const int tid = threadIdx.x;
  const int wave = tid >> 5;
  const int lane = tid & 31;
  const int Hh = (lane >> 4) & 1;
  const int l16 = lane & 15;

  const float lnL = logL[0];
  const float Lf = expf(lnL);
  const float Gamma = expf(logGamma[0]);
  const float Mf = expf(logM[0]);
  const float L2 = Lf * Lf;
  const float pref = Mf * expf((5.f / 3.f) * lnL);  // M * L^(5/3)
  const float k1v = k1[i1];

  // --- load MLP constants once (wave-uniform A operands / per-lane slices) ---
  MLP P;
  // layer0: Wᵀ A-layout, feature K = 2*Hh+v, rows >=3 padded with 0
  P.a0.x = W0[(2 * Hh + 0) * 16 + l16];          // K = 0 or 2 (always < 3)
  P.a0.y = Hh ? 0.f : W0[1 * 16 + l16];          // K = 1, or padded K=3 -> 0
#pragma unroll
  for (int c = 0; c < 4; ++c) {
    P.a1[c].x = W1[(4 * c + 2 * Hh + 0) * 16 + l16];
    P.a1[c].y = W1[(4 * c + 2 * Hh + 1) * 16 + l16];
    P.a2[c].x = W2[(4 * c + 2 * Hh + 0) * 16 + l16];
    P.a2[c].y = W2[(4 * c + 2 * Hh + 1) * 16 + l16];
  }
#pragma unroll
  for (int r = 0; r < 8; ++r) {
    P.cb0[r] = b0[Hh * 8 + r];
    P.cb1[r] = b1[Hh * 8 + r];
    P.cb2[r] = b2[Hh * 8 + r];
    P.w3v[r] = W3[Hh * 8 + r];
  }
  P.b3s = b3[0];

  float acc0 = 0.f, acc1 = 0.f, acc2 = 0.f, acc3 = 0.f;

  // Block covers 16 rows of i2 x 192 of i3 = 96 groups of 32 points; 12 per wave.
  for (int j = 0; j < 12; ++j) {
    const int g = wave * 12 + j;
    const int i2 = tile * 16 + g / 6;
    const int i3 = (g % 6) * 32 + lane;
    const float k2v = k2[i2];
    const float k3v = k3[i3];
    // composite trapezoid weights (clamped central difference)
    const float w2 = 0.5f * (k2[min(i2 + 1, NK2 - 1)] - k2[max(i2 - 1, 0)]);
    const float w3 = 0.5f * (k3[min(i3 + 1, NK3 - 1)] - k3[max(i3 - 1, 0)]);

    const float kk = k1v * k1v + k2v * k2v + k3v * k3v;
    const float beta = Gamma * expf((-2.f / 3.f) * (lnL + 0.5f * logf(kk)));
    const float k30 = k3v + beta * k1v;
    const float kk0 = k1v * k1v + k2v * k2v + k30 * k30;
    const float kL0sq = L2 * kk0;
    // E0 = M * L^(5/3) * (L*sqrt(kk0))^4 / (1 + L^2*kk0)^(17/6)
    const float E0 =
        pref * expf(2.f * logf(kL0sq) - (17.f / 6.f) * log1pf(kL0sq));

    // --- MLP on (k1,k2,k3): 32 lane-points as two 16-point WMMA tiles ---
    // k1v,k2v are wave-uniform; only k3 needs the cross-half shuffle.
    const float K3x = lx16(k3v);
    v2f b0t0, b0t1;
    b0t0.x = Hh ? K3x : k1v;  b0t0.y = Hh ? 0.f : k2v;  // tile: points = lanes 0-15
    b0t1.x = Hh ? k3v : k1v;  b0t1.y = Hh ? 0.f : k2v;  // tile: points = lanes 16-31
    const float m0 = mlp_tile(b0t0, P, Hh);
    const float m1 = mlp_tile(b0t1, P, Hh);
    const float mlp_out = (lane < 16) ? m0 : m1;

    const float C3 = (0.1f + mlp_out) / (1.f + kk);
    const float s2 = k1v * k1v + k2v * k2v;
    const float sq = sqrtf(s2);
    const float C1 =
        beta * k1v * k1v * (kk0 - 2.f * k30 * k30 + beta * k1v * k30) / (kk * s2);
    const float C2 =
        k2v * kk0 / (s2 * sq) * (atanf(k30 / sq) - atanf(k3v / sq));
    const float rr = k2v / k1v;
    const float z1 = C1 - rr * C2 - rr * C3;
    const float z2 = rr * C1 + C2 + C3;
    const float ikk0 = 1.f / kk0;
    const float ikk = 1.f / kk;
    const float Phi11 =
        E0 * ikk0 * ikk0 * (kk0 - k1v * k1v - 2.f * k1v * k30 * z1 + s2 * z1 * z1);
    const float Phi22 =
        E0 * ikk0 * ikk0 * (kk0 - k2v * k2v - 2.f * k2v * k30 * z2 + s2 * z2 * z2);
    const float Phi33 = E0 * ikk * ikk * s2;
    const float Phi13 = E0 * ikk * ikk0 * (-k1v * k30 + s2 * z1);

    const float w = w2 * w3;
    acc0 += w * Phi11;
    acc1 += w * Phi22;
    acc2 += w * Phi33;
    acc3 += w * Phi13;
  }

  // --- deterministic block reduction ---
  __shared__ float sred[8][4];
  float a[4] = {acc0, acc1, acc2, acc3};
#pragma unroll
  for (int c = 0; c < 4; ++c) {
    float v = a[c];
#pragma unroll
    for (int off = 16; off >= 1; off >>= 1) v += __shfl_xor(v, off, 32);
    if (lane == 0) sred[wave][c] = v;
  }
  __syncthreads();
  if (tid < 4) {
    float s = 0.f;
#pragma unroll
    for (int wv = 0; wv < 8; ++wv) s += sred[wv][tid];
    partial[(tid * NK1 + i1) * NT2 + tile] = s;
  }
}

__global__ void __launch_bounds__(256) ops_phase2(
    const float* __restrict__ k1, const float* __restrict__ partial,
    float* __restrict__ out) {
  const int t = threadIdx.x;   // 256 threads: c = t/64, i1 = t%64
  const int c = t >> 6;
  const int i1 = t & 63;
  float s = 0.f;
#pragma unroll
  for (int tl = 0; tl < NT2; ++tl) s += partial[(c * NK1 + i1) * NT2 + tl];
  out[c * NK1 + i1] = 4.f * k1[i1] * s;
}

extern "C" void kernel_launch(void* const* d_in, const int* in_sizes, int n_in,
                              void* d_out, int out_size, void* d_ws, size_t ws_size,
                              hipStream_t stream) {
  (void)in_sizes; (void)n_in; (void)out_size; (void)ws_size;
  const float* k1 = (const float*)d_in[0];
  const float* k2 = (const float*)d_in[1];
  const float* k3 = (const float*)d_in[2];
  const float* logL = (const float*)d_in[3];
  const float* logGamma = (const float*)d_in[4];
  const float* logM = (const float*)d_in[5];
  const float* W0 = (const float*)d_in[6];
  const float* b0 = (const float*)d_in[7];
  const float* W1 = (const float*)d_in[8];
  const float* b1 = (const float*)d_in[9];
  const float* W2 = (const float*)d_in[10];
  const float* b2 = (const float*)d_in[11];
  const float* W3 = (const float*)d_in[12];
  const float* b3 = (const float*)d_in[13];

  float* partial = (float*)d_ws;  // 4*64*12 floats = 12 KB
  float* out = (float*)d_out;

  ops_phase1<<<dim3(NK1, NT2), 256, 0, stream>>>(
      k1, k2, k3, logL, logGamma, logM, W0, b0, W1, b1, W2, b2, W3, b3, partial);
  ops_phase2<<<1, 256, 0, stream>>>(k1, partial, out);
}